// TopKGate_14783277433021
// MI455X (gfx1250) — compile-verified
//
#include <hip/hip_runtime.h>
#include <hip/hip_bf16.h>
#include <float.h>

// Problem constants (from reference)
#define S_TOK   8192
#define M_DIM   2048
#define GE_EXP  64
#define K_TOP   2
#define CAPACITY 256

typedef float v2f __attribute__((ext_vector_type(2)));
typedef float v8f __attribute__((ext_vector_type(8)));

// ---------------------------------------------------------------------------
// Kernel 1: router logits = x @ wg.T  via V_WMMA_F32_16X16X4_F32 (fp32 WMMA).
// One wave handles a 16-row tile and all 64 expert columns (4 accumulators).
// A fragment (16x4 f32): lane L(0..15) row=L, VGPR v holds K = v + (L>=16? 2:0)
//   -> each lane loads float2 at column k + (lane>>4)*2.
// B fragment mirrors A with N striped across lanes.
// C/D (16x16 f32): VGPR r -> row = r + 8*(lane>>4), col = lane&15.
// ---------------------------------------------------------------------------
__global__ __launch_bounds__(256) void moe_logits_wmma(
    const float* __restrict__ x, const float* __restrict__ wg,
    float* __restrict__ logits) {
  const int wave = threadIdx.x >> 5;           // 0..7
  const int lane = threadIdx.x & 31;
  const int mrow = lane & 15;
  const int half = lane >> 4;                  // 0 or 1
  const int rowBase = blockIdx.x * 128 + wave * 16;

  const float* __restrict__ xrow = x + (size_t)(rowBase + mrow) * M_DIM;
  const float* __restrict__ b0p = wg + (size_t)( 0 + mrow) * M_DIM;
  const float* __restrict__ b1p = wg + (size_t)(16 + mrow) * M_DIM;
  const float* __restrict__ b2p = wg + (size_t)(32 + mrow) * M_DIM;
  const float* __restrict__ b3p = wg + (size_t)(48 + mrow) * M_DIM;

  v8f acc0 = {}, acc1 = {}, acc2 = {}, acc3 = {};

  for (int k = 0; k < M_DIM; k += 4) {
    const int ka = k + half * 2;
    // hint the next x cacheline into the hierarchy
    __builtin_prefetch(xrow + ka + 32, 0, 0);
    v2f a  = *(const v2f*)(xrow + ka);
    v2f b0 = *(const v2f*)(b0p + ka);
    v2f b1 = *(const v2f*)(b1p + ka);
    v2f b2 = *(const v2f*)(b2p + ka);
    v2f b3 = *(const v2f*)(b3p + ka);
    acc0 = __builtin_amdgcn_wmma_f32_16x16x4_f32(false, a, false, b0,
                                                 (short)0, acc0, false, false);
    acc1 = __builtin_amdgcn_wmma_f32_16x16x4_f32(false, a, false, b1,
                                                 (short)0, acc1, false, false);
    acc2 = __builtin_amdgcn_wmma_f32_16x16x4_f32(false, a, false, b2,
                                                 (short)0, acc2, false, false);
    acc3 = __builtin_amdgcn_wmma_f32_16x16x4_f32(false, a, false, b3,
                                                 (short)0, acc3, false, false);
  }

  // Write back: VGPR r -> row rowBase + r + 8*half, col = tile*16 + mrow
  float* __restrict__ out = logits + (size_t)(rowBase + half * 8) * GE_EXP + mrow;
#pragma unroll
  for (int r = 0; r < 8; ++r) {
    float* orow = out + (size_t)r * GE_EXP;
    orow[0]  = acc0[r];
    orow[16] = acc1[r];
    orow[32] = acc2[r];
    orow[48] = acc3[r];
  }
}

// ---------------------------------------------------------------------------
// Kernel 2: per-token softmax over 64 experts, top-2 selection, normalized
// top-k gates. Stores the full softmax row (needed for the aux-loss "me").
// ---------------------------------------------------------------------------
__global__ __launch_bounds__(256) void moe_gate(
    const float* __restrict__ logits, float* __restrict__ gates,
    int* __restrict__ topi, float* __restrict__ gval) {
  const int s = blockIdx.x * blockDim.x + threadIdx.x;
  if (s >= S_TOK) return;
  const float* __restrict__ lrow = logits + (size_t)s * GE_EXP;

  float mx = -FLT_MAX;
  float v0 = -FLT_MAX, v1 = -FLT_MAX;
  int i0 = 0, i1 = 0;
#pragma unroll 8
  for (int e = 0; e < GE_EXP; ++e) {
    float l = lrow[e];
    mx = fmaxf(mx, l);
    if (l > v0) { v1 = v0; i1 = i0; v0 = l; i0 = e; }
    else if (l > v1) { v1 = l; i1 = e; }
  }
  float sum = 0.f;
  float* __restrict__ grow = gates + (size_t)s * GE_EXP;
#pragma unroll 8
  for (int e = 0; e < GE_EXP; ++e) {
    float g = expf(lrow[e] - mx);
    grow[e] = g;
    sum += g;
  }
  float inv = 1.f / sum;
#pragma unroll 8
  for (int e = 0; e < GE_EXP; ++e) grow[e] *= inv;

  float g0 = expf(v0 - mx) * inv;
  float g1 = expf(v1 - mx) * inv;
  float denom = fmaxf(g0 + g1, 1.1920928955078125e-07f);  // fp32 eps clamp
  gval[s * K_TOP + 0] = g0 / denom;
  gval[s * K_TOP + 1] = g1 / denom;
  topi[s * K_TOP + 0] = i0;
  topi[s * K_TOP + 1] = i1;
}

// ---------------------------------------------------------------------------
// Kernel 3: deterministic capacity assignment. One thread per expert walks
// tokens in order (matches the reference's cumsum-over-S semantics exactly):
//   k=0 pass assigns 0..cnt0-1; k=1 pass starts at counts0[e] (acc_base).
// Also computes me[e] = sum_s gates[s,e], ce[e] = counts0[e] and the
// load-balance loss, reduced in LDS.
// ---------------------------------------------------------------------------
__global__ __launch_bounds__(64) void moe_assign(
    const int* __restrict__ topi, const float* __restrict__ gates,
    int* __restrict__ loc, float* __restrict__ lloss_out) {
  __shared__ int s_c0[GE_EXP];
  __shared__ float s_red[GE_EXP];
  const int e = threadIdx.x;

  int cnt = 0;
  for (int s = 0; s < S_TOK; ++s)
    if (topi[s * K_TOP + 0] == e) loc[s * K_TOP + 0] = cnt++;
  s_c0[e] = cnt;
  __syncthreads();

  int cnt1 = s_c0[e];  // exclusive prefix over k for this expert
  for (int s = 0; s < S_TOK; ++s)
    if (topi[s * K_TOP + 1] == e) loc[s * K_TOP + 1] = cnt1++;

  float me = 0.f;
  for (int s = 0; s < S_TOK; ++s) me += gates[(size_t)s * GE_EXP + e];
  s_red[e] = me * (float)s_c0[e];
  __syncthreads();

  for (int off = 32; off > 0; off >>= 1) {
    if (e < off) s_red[e] += s_red[e + off];
    __syncthreads();
  }
  if (e == 0)
    lloss_out[0] = s_red[0] * ((float)GE_EXP / ((float)S_TOK * (float)S_TOK));
}

// ---------------------------------------------------------------------------
// Kernel 4: combine. Identity expert + unique slots => y[s] = scale[s]*x[s],
// scale = sum_k gate_k * (loc_k < CAPACITY). Pure bandwidth: b128 load/store.
// ---------------------------------------------------------------------------
__global__ __launch_bounds__(256) void moe_combine(
    const float* __restrict__ x, const float* __restrict__ gval,
    const int* __restrict__ loc, float* __restrict__ y) {
  const int s = blockIdx.x;
  const float w0 = gval[s * K_TOP + 0] * ((loc[s * K_TOP + 0] < CAPACITY) ? 1.f : 0.f);
  const float w1 = gval[s * K_TOP + 1] * ((loc[s * K_TOP + 1] < CAPACITY) ? 1.f : 0.f);
  const float scale = w0 + w1;

  const float4* __restrict__ xr = (const float4*)(x + (size_t)s * M_DIM);
  float4* __restrict__ yr = (float4*)(y + (size_t)s * M_DIM);
#pragma unroll 2
  for (int j = threadIdx.x; j < M_DIM / 4; j += 256) {
    float4 v = xr[j];
    v.x *= scale; v.y *= scale; v.z *= scale; v.w *= scale;
    yr[j] = v;
  }
}

// ---------------------------------------------------------------------------
extern "C" void kernel_launch(void* const* d_in, const int* in_sizes, int n_in,
                              void* d_out, int out_size, void* d_ws, size_t ws_size,
                              hipStream_t stream) {
  (void)in_sizes; (void)n_in; (void)out_size; (void)ws_size;
  const float* x  = (const float*)d_in[0];   // [S, M]
  const float* wg = (const float*)d_in[1];   // [GE, M]
  float* y = (float*)d_out;                  // [S, M] + 1 (l_loss)
  float* lloss = y + (size_t)S_TOK * M_DIM;

  // workspace layout
  float* logits = (float*)d_ws;                        // S*GE
  float* gates  = logits + (size_t)S_TOK * GE_EXP;     // S*GE
  int*   topi   = (int*)(gates + (size_t)S_TOK * GE_EXP); // S*K
  float* gval   = (float*)(topi + (size_t)S_TOK * K_TOP); // S*K
  int*   loc    = (int*)(gval + (size_t)S_TOK * K_TOP);   // S*K

  moe_logits_wmma<<<S_TOK / 128, 256, 0, stream>>>(x, wg, logits);
  moe_gate<<<S_TOK / 256, 256, 0, stream>>>(logits, gates, topi, gval);
  moe_assign<<<1, 64, 0, stream>>>(topi, gates, loc, lloss);
  moe_combine<<<S_TOK, 256, 0, stream>>>(x, gval, loc, y);
}